// CompositeLoss_59554016526632
// MI455X (gfx1250) — compile-verified
//
#include <hip/hip_runtime.h>
#include <hip/hip_bf16.h>

// ---------------------------------------------------------------------------
// CompositeLoss for MI455X (gfx1250): L1 + SSIM(7x7x7 box) + gradient loss.
// Bandwidth-bound (~151 MB working set, 23.3 TB/s HBM, fits in 192 MB L2).
// SSIM box filters are done as banded-matrix WMMAs (V_WMMA_F32_16X16X4_F32)
// entirely in LDS per 16^3 tile; all reductions are deterministic partials.
// ---------------------------------------------------------------------------

typedef __attribute__((ext_vector_type(2))) float v2f;
typedef __attribute__((ext_vector_type(8))) float v8f;

#define NB    2
#define DD    128
#define HH    192
#define WW    192
#define PLANE (HH * WW)            // 36864
#define VOL   (DD * HH * WW)       // 4718592
#define NTOT  (NB * VOL)           // 9437184
#define NBLK  2304                 // 2 * (128/16) * (192/16) * (192/16)
#define NTHR  256
#define TL    22                   // 16 + 2*3 halo
#define T3    (TL * TL * TL)       // 10648

#define C1F   (0.01f * 0.01f)
#define C2F   (0.03f * 0.03f)

static __device__ __forceinline__ int imin(int a, int b) { return a < b ? a : b; }

static __device__ __forceinline__ float block_reduce(float v, float* red) {
  const int tid = threadIdx.x;
  red[tid] = v;
  __syncthreads();
  #pragma unroll
  for (int s = NTHR / 2; s > 0; s >>= 1) {
    if (tid < s) red[tid] += red[tid + s];
    __syncthreads();
  }
  float r = red[0];
  __syncthreads();
  return r;
}

// ---------------------------------------------------------------------------
// Kernel 1: L1 + gradient loss.  fdiff is linear, so
// fdiff(pred)-fdiff(tgt) == fdiff(pred-tgt); one fused streaming pass.
// ---------------------------------------------------------------------------
__global__ __launch_bounds__(NTHR)
void grad_l1_kernel(const float* __restrict__ pred, const float* __restrict__ tgt,
                    float* __restrict__ part) {
  const int tid  = threadIdx.x;
  const int gtid = blockIdx.x * NTHR + tid;
  const int STRIDE = NBLK * NTHR;           // 589824; *16 == NTOT exactly

  float sl1 = 0.f, sgx = 0.f, sgy = 0.f, sgz = 0.f;

  #pragma unroll 4
  for (int i = 0; i < 16; ++i) {
    const int idx = gtid + i * STRIDE;
    const int x  = idx % WW;
    const int t1 = idx / WW;
    const int y  = t1 % HH;
    const int t2 = t1 / HH;
    const int z  = t2 % DD;

    const float d0 = pred[idx] - tgt[idx];
    sl1 += fabsf(d0);

    float fd;
    // x axis (stride 1)
    if (x == 0)            fd = (pred[idx + 1] - tgt[idx + 1]) - d0;
    else if (x == WW - 1)  fd = d0 - (pred[idx - 1] - tgt[idx - 1]);
    else                   fd = 0.5f * ((pred[idx + 1] - tgt[idx + 1]) -
                                        (pred[idx - 1] - tgt[idx - 1]));
    sgx += fabsf(fd);

    // y axis (stride WW)
    if (y == 0)            fd = (pred[idx + WW] - tgt[idx + WW]) - d0;
    else if (y == HH - 1)  fd = d0 - (pred[idx - WW] - tgt[idx - WW]);
    else                   fd = 0.5f * ((pred[idx + WW] - tgt[idx + WW]) -
                                        (pred[idx - WW] - tgt[idx - WW]));
    sgy += fabsf(fd);

    // z axis (stride PLANE), per-sample edges
    if (z == 0)            fd = (pred[idx + PLANE] - tgt[idx + PLANE]) - d0;
    else if (z == DD - 1)  fd = d0 - (pred[idx - PLANE] - tgt[idx - PLANE]);
    else                   fd = 0.5f * ((pred[idx + PLANE] - tgt[idx + PLANE]) -
                                        (pred[idx - PLANE] - tgt[idx - PLANE]));
    sgz += fabsf(fd);
  }

  __shared__ float red[NTHR];
  float r0 = block_reduce(sl1, red);
  float r1 = block_reduce(sgx, red);
  float r2 = block_reduce(sgy, red);
  float r3 = block_reduce(sgz, red);
  if (tid == 0) {
    part[0 * NBLK + blockIdx.x] = r0;
    part[1 * NBLK + blockIdx.x] = r1;
    part[2 * NBLK + blockIdx.x] = r2;
    part[3 * NBLK + blockIdx.x] = r3;
  }
}

// ---------------------------------------------------------------------------
// Kernel 2: SSIM. One 16^3 output tile per block.  Five fields
// (p, t, p^2, t^2, p*t) are box-filtered 7x7x7 separably:
//   X pass: register sliding-window sum (VALU),
//   Y pass: Out(16y x 16x) = Band(16x22) x Data(22x16)  via 6x WMMA f32 16x16x4,
//   Z pass: same banded WMMA over z.
// Band[i][j] = 1 iff i <= j <= i+6 (halo 3 each side, zero padded).
// A-fragment layout per ISA: lane L holds row M=L%16; v0 = K(0|2), v1 = K(1|3).
// B assumed symmetric: v0 = rows K(0|2) with N = lane%16, v1 = rows K(1|3).
// C/D layout: lane L, vgpr r -> M = r + (L>=16 ? 8 : 0), N = L%16.
// ---------------------------------------------------------------------------
__global__ __launch_bounds__(NTHR)
void ssim_kernel(const float* __restrict__ pred, const float* __restrict__ tgt,
                 float* __restrict__ part) {
  __shared__ float pS[T3];                 // clipped pred tile, 22^3
  __shared__ float tS[T3];                 // clipped tgt  tile, 22^3
  __shared__ float Ax[TL * TL * 16];       // X-filtered: [z][y][x']
  __shared__ float By[TL * 16 * 16];       // XY-filtered: [z][y'][x']
  __shared__ float red[NTHR];

  const int tid  = threadIdx.x;
  const int lane = tid & 31;
  const int wave = tid >> 5;               // 8 waves

  const int bt = blockIdx.x;
  const int xt = bt % 12;
  const int yt = (bt / 12) % 12;
  const int zt = (bt / 144) % 8;
  const int b  = bt / 1152;
  const int x0 = xt * 16, y0 = yt * 16, z0 = zt * 16;
  const float* __restrict__ pb = pred + (long long)b * VOL;
  const float* __restrict__ tb = tgt  + (long long)b * VOL;

  // ---- load halo tiles, clip to [0,1], zero-pad outside the volume ----
  for (int i = tid; i < T3; i += NTHR) {
    const int lx = i % TL;
    const int rr = i / TL;
    const int ly = rr % TL;
    const int lz = rr / TL;
    const int gx = x0 + lx - 3, gy = y0 + ly - 3, gz = z0 + lz - 3;
    float pv = 0.f, tv = 0.f;
    if (gx >= 0 && gx < WW && gy >= 0 && gy < HH && gz >= 0 && gz < DD) {
      const int g = (gz * HH + gy) * WW + gx;
      pv = fminf(fmaxf(pb[g], 0.f), 1.f);
      tv = fminf(fmaxf(tb[g], 0.f), 1.f);
    }
    pS[i] = pv;
    tS[i] = tv;
  }
  __syncthreads();

  // ---- per-lane banded A fragments (same band for Y and Z passes) ----
  v2f band[6];
  {
    const int m   = lane & 15;
    const int kof = (lane < 16) ? 0 : 2;
    #pragma unroll
    for (int c = 0; c < 6; ++c) {
      const int j0 = 4 * c + kof;
      band[c].x = (j0     >= m && j0     <= m + 6 && j0     < TL) ? 1.f : 0.f;
      band[c].y = (j0 + 1 >= m && j0 + 1 <= m + 6 && j0 + 1 < TL) ? 1.f : 0.f;
    }
  }

  const int n   = lane & 15;
  const int kof = (lane < 16) ? 0 : 2;
  const int mb  = (lane < 16) ? 0 : 8;     // M offset of this lane in C/D tile

  v8f res[5][2];                           // 5 fields x 2 y-slices per wave

  #pragma unroll
  for (int f = 0; f < 5; ++f) {
    // ---- X pass: sliding 7-tap sum along x for all (z,y) rows ----
    for (int r = tid; r < TL * TL; r += NTHR) {
      const float* prow = &pS[r * TL];
      const float* trow = &tS[r * TL];
      float v[TL];
      #pragma unroll
      for (int x = 0; x < TL; ++x) {
        const float pv = prow[x], tv = trow[x];
        v[x] = (f == 0) ? pv
             : (f == 1) ? tv
             : (f == 2) ? pv * pv
             : (f == 3) ? tv * tv
             :            pv * tv;
      }
      float s = v[0] + v[1] + v[2] + v[3] + v[4] + v[5] + v[6];
      float* orow = &Ax[r * 16];
      orow[0] = s;
      #pragma unroll
      for (int x = 1; x < 16; ++x) { s += v[x + 6] - v[x - 1]; orow[x] = s; }
    }
    __syncthreads();

    // ---- Y pass (WMMA): for each z slice, Out = Band x Ax[z] ----
    for (int z = wave; z < TL; z += 8) {   // uniform per-wave trip count
      v8f acc = {0.f, 0.f, 0.f, 0.f, 0.f, 0.f, 0.f, 0.f};
      #pragma unroll
      for (int c = 0; c < 6; ++c) {
        const int j0 = 4 * c + kof;
        const int ja = imin(j0,     TL - 1);   // band is 0 beyond j>=22
        const int jb = imin(j0 + 1, TL - 1);
        v2f bf;
        bf.x = Ax[(z * TL + ja) * 16 + n];
        bf.y = Ax[(z * TL + jb) * 16 + n];
        acc = __builtin_amdgcn_wmma_f32_16x16x4_f32(
                false, band[c], false, bf, (short)0, acc, false, false);
      }
      #pragma unroll
      for (int r = 0; r < 8; ++r)
        By[(z * 16 + (r + mb)) * 16 + n] = acc[r];
    }
    __syncthreads();

    // ---- Z pass (WMMA): for this wave's y slices, Out = Band x By[:,y,:] ----
    #pragma unroll
    for (int s2 = 0; s2 < 2; ++s2) {
      const int y = wave + 8 * s2;
      v8f acc = {0.f, 0.f, 0.f, 0.f, 0.f, 0.f, 0.f, 0.f};
      #pragma unroll
      for (int c = 0; c < 6; ++c) {
        const int j0 = 4 * c + kof;
        const int ja = imin(j0,     TL - 1);
        const int jb = imin(j0 + 1, TL - 1);
        v2f bf;
        bf.x = By[(ja * 16 + y) * 16 + n];
        bf.y = By[(jb * 16 + y) * 16 + n];
        acc = __builtin_amdgcn_wmma_f32_16x16x4_f32(
                false, band[c], false, bf, (short)0, acc, false, false);
      }
      res[f][s2] = acc;
    }
    __syncthreads();   // before next field overwrites Ax/By
  }

  // ---- SSIM formula per voxel; each wave owns y = wave, wave+8 slices ----
  const float inv = 1.f / 343.f;
  float ssum = 0.f;
  #pragma unroll
  for (int s2 = 0; s2 < 2; ++s2) {
    #pragma unroll
    for (int e = 0; e < 8; ++e) {
      const float mp  = res[0][s2][e] * inv;
      const float mt  = res[1][s2][e] * inv;
      const float ep2 = res[2][s2][e] * inv;
      const float et2 = res[3][s2][e] * inv;
      const float ept = res[4][s2][e] * inv;
      const float mp2 = mp * mp, mt2 = mt * mt, mpt = mp * mt;
      const float sp  = fmaxf(ep2 - mp2, 0.f);
      const float st  = fmaxf(et2 - mt2, 0.f);
      const float spt = ept - mpt;
      const float num = (2.f * mpt + C1F) * (2.f * spt + C2F);
      const float den = (mp2 + mt2 + C1F) * (sp + st + C2F) + 1e-8f;
      ssum += num / den;
    }
  }
  const float tot = block_reduce(ssum, red);
  if (tid == 0) part[blockIdx.x] = tot;    // part == ws + 4*NBLK
}

// ---------------------------------------------------------------------------
// Kernel 3: combine partials into the scalar loss.
// ---------------------------------------------------------------------------
__global__ __launch_bounds__(NTHR)
void finalize_kernel(const float* __restrict__ part, float* __restrict__ out) {
  __shared__ float red[NTHR];
  const int tid = threadIdx.x;
  float s[5] = {0.f, 0.f, 0.f, 0.f, 0.f};
  for (int i = tid; i < NBLK; i += NTHR) {
    s[0] += part[0 * NBLK + i];   // l1 sum
    s[1] += part[1 * NBLK + i];   // |fdiff_x|
    s[2] += part[2 * NBLK + i];   // |fdiff_y|
    s[3] += part[3 * NBLK + i];   // |fdiff_z|
    s[4] += part[4 * NBLK + i];   // ssim ratio sum
  }
  float t[5];
  #pragma unroll
  for (int k = 0; k < 5; ++k) t[k] = block_reduce(s[k], red);
  if (tid == 0) {
    const float Nf   = (float)NTOT;          // 9437184 (exactly representable)
    const float l1   = t[0] / Nf;
    const float grad = (t[1] + t[2] + t[3]) / (3.f * Nf);
    const float ssiml = 1.f - t[4] / Nf;
    out[0] = 0.7f * l1 + 0.2f * ssiml + 0.1f * grad;
  }
}

// ---------------------------------------------------------------------------
extern "C" void kernel_launch(void* const* d_in, const int* in_sizes, int n_in,
                              void* d_out, int out_size, void* d_ws, size_t ws_size,
                              hipStream_t stream) {
  const float* pred = (const float*)d_in[0];
  const float* tgt  = (const float*)d_in[1];
  float* ws  = (float*)d_ws;            // layout: [5][NBLK] partial sums
  float* out = (float*)d_out;

  grad_l1_kernel<<<NBLK, NTHR, 0, stream>>>(pred, tgt, ws);
  ssim_kernel<<<NBLK, NTHR, 0, stream>>>(pred, tgt, ws + 4 * NBLK);
  finalize_kernel<<<1, NTHR, 0, stream>>>(ws, out);
}